// BaseGNN_25297357373591
// MI455X (gfx1250) — compile-verified
//
#include <hip/hip_runtime.h>
#include <hip/hip_bf16.h>

typedef float v2f __attribute__((ext_vector_type(2)));
typedef float v8f __attribute__((ext_vector_type(8)));

#define C 64
#define NEG_SLOPE 0.01f

__device__ __forceinline__ void atomic_add_f32(float* p, float v) {
    // relaxed, device scope -> GLOBAL_ATOMIC_ADD_F32 (no return)
    __hip_atomic_fetch_add(p, v, __ATOMIC_RELAXED, __HIP_MEMORY_SCOPE_AGENT);
}

// ---------------- degree accumulation ----------------
__global__ void deg_kernel(const int* __restrict__ src, const int* __restrict__ dst,
                           float* __restrict__ degO, float* __restrict__ degI, int E) {
    int e = blockIdx.x * blockDim.x + threadIdx.x;
    if (e < E) {
        atomic_add_f32(&degO[src[e]], 1.0f);
        atomic_add_f32(&degI[dst[e]], 1.0f);
    }
}

// deg -> clamped inverse sqrt, in place
__global__ void invsqrt_kernel(float* __restrict__ degO, float* __restrict__ degI, int N) {
    int i = blockIdx.x * blockDim.x + threadIdx.x;
    if (i < N) {
        degO[i] = rsqrtf(fmaxf(degO[i], 1.0f));
        degI[i] = rsqrtf(fmaxf(degI[i], 1.0f));
    }
}

// ---------------- edge scatter: agg[dst] += feat[src] * oinv[src] ----------------
// one thread per (edge, channel-pair): float2 gather, two f32 atomic adds.
// 32 consecutive threads share an edge -> src/dst/oinv loads broadcast in-wave.
__global__ void scatter_kernel(const int* __restrict__ src, const int* __restrict__ dst,
                               const float* __restrict__ feat, const float* __restrict__ oinv,
                               float* __restrict__ agg, long total2) {
    long idx = (long)blockIdx.x * blockDim.x + threadIdx.x;
    if (idx < total2) {
        int e = (int)(idx >> 5);
        int c = (int)(idx & 31) * 2;
        int s = src[e];
        int d = dst[e];
        float sc = oinv[s];
        v2f f = *(const v2f*)(feat + (long)s * C + c);
        float* out = agg + (long)d * C + c;
        atomic_add_f32(out + 0, f.x * sc);
        atomic_add_f32(out + 1, f.y * sc);
    }
}

// ---------------- WMMA GEMM: h = leaky_relu((agg * iinv[:,None]) @ W + b) ----------------
// block = 256 threads = 8 waves; wave -> (rowTile 0..1, colTile 0..3); 32 rows per block.
// V_WMMA_F32_16X16X4_F32: fp32 A/B, fp32 accum -- exact fp32 math.
__global__ void gemm_wmma_kernel(const float* __restrict__ agg, const float* __restrict__ iinv,
                                 const float* __restrict__ W, const float* __restrict__ bias,
                                 float* __restrict__ h) {
    int lane = threadIdx.x & 31;
    int wave = threadIdx.x >> 5;           // 0..7
    int rowTile = wave >> 2;               // 0..1
    int colTile = wave & 3;                // 0..3
    int rowBase = blockIdx.x * 32 + rowTile * 16;
    int colBase = colTile * 16;
    int half = lane >> 4;                  // 0: K {0,1}; 1: K {2,3}
    int l15  = lane & 15;

    v8f acc = {};
    const float* arow = agg + (long)(rowBase + l15) * C;   // A row (M = l15)

#pragma unroll
    for (int k0 = 0; k0 < C; k0 += 4) {
        int ka = k0 + half * 2;
        // A 16x4 tile: lane holds a[M=l15][ka..ka+1] -> contiguous float2
        v2f a = *(const v2f*)(arow + ka);
        // B 4x16 tile: lane holds W[ka][colBase+l15], W[ka+1][colBase+l15]
        v2f b;
        b.x = W[(long)ka * C + colBase + l15];
        b.y = W[(long)(ka + 1) * C + colBase + l15];
        acc = __builtin_amdgcn_wmma_f32_16x16x4_f32(
            /*neg_a=*/false, a, /*neg_b=*/false, b,
            /*c_mod=*/(short)0, acc, /*reuse_a=*/false, /*reuse_b=*/false);
    }

    // D layout: VGPR j -> M = j + 8*half, N = l15
#pragma unroll
    for (int j = 0; j < 8; ++j) {
        int row = rowBase + j + 8 * half;
        int col = colBase + l15;
        float v = acc[j] * iinv[row] + bias[col];
        h[(long)row * C + col] = (v > 0.0f) ? v : NEG_SLOPE * v;
    }
}

// scalar tail (only launched if N % 32 != 0; N=100000 -> unused, kept for generality)
__global__ void gemm_tail_kernel(const float* __restrict__ agg, const float* __restrict__ iinv,
                                 const float* __restrict__ W, const float* __restrict__ bias,
                                 float* __restrict__ h, int rowStart, int nRows) {
    int idx = blockIdx.x * blockDim.x + threadIdx.x;
    if (idx < nRows * C) {
        int row = rowStart + idx / C;
        int col = idx % C;
        float s = 0.0f;
        for (int k = 0; k < C; ++k) s += agg[(long)row * C + k] * W[(long)k * C + col];
        float v = s * iinv[row] + bias[col];
        h[(long)row * C + col] = (v > 0.0f) ? v : NEG_SLOPE * v;
    }
}

// ---------------- head: out = mean(h[:order+1]) @ Wl + bl ----------------
__global__ void head_kernel(const float* __restrict__ h, const float* __restrict__ Wl,
                            const float* __restrict__ bl, const int* __restrict__ orderPtr,
                            float* __restrict__ out) {
    __shared__ float pooled[C];
    int c = threadIdx.x;     // 0..63
    int rows = *orderPtr + 1;
    float s = 0.0f;
    for (int i = 0; i < rows; ++i) s += h[(long)i * C + c];
    pooled[c] = s / (float)rows;
    __syncthreads();
    float acc = bl[c];
    for (int k = 0; k < C; ++k) acc += pooled[k] * Wl[(long)k * C + c];
    out[c] = acc;
}

extern "C" void kernel_launch(void* const* d_in, const int* in_sizes, int n_in,
                              void* d_out, int out_size, void* d_ws, size_t ws_size,
                              hipStream_t stream) {
    const int*   src   = (const int*)d_in[0];
    const int*   dst   = (const int*)d_in[1];
    const float* feat  = (const float*)d_in[2];
    const float* W1    = (const float*)d_in[3];
    const float* b1    = (const float*)d_in[4];
    const float* W2    = (const float*)d_in[5];
    const float* b2    = (const float*)d_in[6];
    const float* Wl    = (const float*)d_in[7];
    const float* bl    = (const float*)d_in[8];
    const int*   order = (const int*)d_in[9];

    int E = in_sizes[0];
    int N = in_sizes[2] / C;

    float* ws   = (float*)d_ws;
    float* degO = ws;                       // N floats (becomes out_deg^-0.5)
    float* degI = ws + N;                   // N floats (becomes in_deg^-0.5)
    float* agg  = ws + 2 * (size_t)N;       // N*C floats
    float* h    = agg + (size_t)N * C;      // N*C floats

    // degrees
    hipMemsetAsync(degO, 0, 2 * (size_t)N * sizeof(float), stream);
    deg_kernel<<<(E + 255) / 256, 256, 0, stream>>>(src, dst, degO, degI, E);
    invsqrt_kernel<<<(N + 255) / 256, 256, 0, stream>>>(degO, degI, N);

    long total2  = (long)E * (C / 2);       // thread per channel-pair
    int  sblocks = (int)((total2 + 255) / 256);
    int  nFull   = N / 32;                  // WMMA blocks (32 rows each)
    int  rem     = N - nFull * 32;

    // ---- layer 1 ----
    hipMemsetAsync(agg, 0, (size_t)N * C * sizeof(float), stream);
    scatter_kernel<<<sblocks, 256, 0, stream>>>(src, dst, feat, degO, agg, total2);
    gemm_wmma_kernel<<<nFull, 256, 0, stream>>>(agg, degI, W1, b1, h);
    if (rem)
        gemm_tail_kernel<<<(rem * C + 255) / 256, 256, 0, stream>>>(agg, degI, W1, b1, h,
                                                                    nFull * 32, rem);

    // ---- layer 2 ----
    hipMemsetAsync(agg, 0, (size_t)N * C * sizeof(float), stream);
    scatter_kernel<<<sblocks, 256, 0, stream>>>(src, dst, h, degO, agg, total2);
    gemm_wmma_kernel<<<nFull, 256, 0, stream>>>(agg, degI, W2, b2, h);
    if (rem)
        gemm_tail_kernel<<<(rem * C + 255) / 256, 256, 0, stream>>>(agg, degI, W2, b2, h,
                                                                    nFull * 32, rem);

    // ---- head ----
    head_kernel<<<1, C, 0, stream>>>(h, Wl, bl, order, (float*)d_out);
}